// HashEncoder_37787122270415
// MI455X (gfx1250) — compile-verified
//
#include <hip/hip_runtime.h>
#include <hip/hip_bf16.h>
#include <math.h>

#define L_LEVELS 16
#define T_SIZE   16384
#define F_FEAT   2
#define NPTS     262144
#define CHUNKS   64
#define PTS_PER_BLOCK (NPTS / CHUNKS)   // 4096
#define THREADS  256

typedef __attribute__((ext_vector_type(4))) unsigned int v4u;
typedef __attribute__((ext_vector_type(8))) int          v8i;
typedef __attribute__((ext_vector_type(4))) int          v4i;

// Issue a TDM DMA of one level's (T,F) f32 table (128 KB, contiguous) from
// global memory into LDS. Descriptor per CDNA5 ISA.
// Group0: count=1 | lds_addr | global_addr[56:0] | type=2.
// Group1: data_size=4B, tensor_dim0=tile_dim0=32768 elems, dim1=1, stride0=32768.
__device__ __forceinline__ void tdm_load_table(const float* gsrc, void* lds_dst) {
    unsigned long long ga = (unsigned long long)(uintptr_t)gsrc;
    unsigned lds_addr = (unsigned)(uintptr_t)lds_dst;   // flat LDS addr low 32 bits = LDS byte offset

    v4u g0;
    g0.x = 1u;                                            // count=1 (valid), gather_mode=0
    g0.y = lds_addr;                                      // lds_addr (bytes)
    g0.z = (unsigned)(ga & 0xFFFFFFFFull);                // global_addr[31:0]
    g0.w = (unsigned)((ga >> 32) & 0x1FFFFFFull)          // global_addr[56:32]
         | 0x80000000u;                                   // type=2 ("image") in bits 127:126

    const unsigned NELEM = T_SIZE * F_FEAT;               // 32768 f32 elements
    v8i g1;
    g1[0] = (int)(2u << 16);                              // workgroup_mask=0, data_size=2 (4 bytes)
    g1[1] = (int)((NELEM & 0xFFFFu) << 16);               // tensor_dim0[15:0] = 32768
    g1[2] = (int)(1u << 16);                              // tensor_dim0[31:16]=0, tensor_dim1=1
    g1[3] = (int)((NELEM & 0xFFFFu) << 16);               // tensor_dim1 hi=0, tile_dim0 = 32768
    g1[4] = 1;                                            // tile_dim1 = 1, tile_dim2 = 0
    g1[5] = (int)NELEM;                                   // tensor_dim0_stride[31:0]
    g1[6] = 0;                                            // stride0 hi, tensor_dim1_stride lo
    g1[7] = 0;

    v4i gz4 = {0, 0, 0, 0};                               // groups 2/3 unused (<=2D tensor)
    v8i gz8 = {0, 0, 0, 0, 0, 0, 0, 0};
    __builtin_amdgcn_tensor_load_to_lds(g0, g1, gz4, gz4, gz8, 0 /*cpol*/);
}

__global__ __launch_bounds__(THREADS) void hash_encode_kernel(
    const float* __restrict__ x,
    const float* __restrict__ emb,
    float* __restrict__ out)
{
    __shared__ float2 tbl[T_SIZE];                        // 128 KB LDS

    const int level = (int)(blockIdx.x & (L_LEVELS - 1));
    const int chunk = (int)(blockIdx.x >> 4);

    // Wave 0 DMAs this level's table into LDS via the Tensor Data Mover.
    if (threadIdx.x < 32) {
        tdm_load_table(emb + (size_t)level * (T_SIZE * F_FEAT), &tbl[0]);
        __builtin_amdgcn_s_wait_tensorcnt(0);
    }
    __syncthreads();

    // N_l = Nmin * b^l with b = exp((ln Nmax - ln Nmin)/L) = 2^(5/16)
    const float scale = (float)exp2(4.0 + 0.3125 * (double)level);

    const unsigned P1 = 2654435761u;                      // wraps like int32 -1640531535
    const unsigned P2 = 805459861u;

    const int base = chunk * PTS_PER_BLOCK;
    #pragma unroll 2
    for (int i = (int)threadIdx.x; i < PTS_PER_BLOCK; i += THREADS) {
        const int n = base + i;
        const float px = x[3 * n + 0] * scale;
        const float py = x[3 * n + 1] * scale;
        const float pz = x[3 * n + 2] * scale;

        const float fx = floorf(px), fy = floorf(py), fz = floorf(pz);
        const float cx = ceilf(px),  cy = ceilf(py),  cz = ceilf(pz);

        // (un - lo)/d with d = ceil-floor in {0,1}: identical to un - floor(un)
        const float dx = px - fx, dy = py - fy, dz = pz - fz;
        const float ex = 1.0f - dx, ey = 1.0f - dy, ez = 1.0f - dz;

        const unsigned hx0 = (unsigned)(int)fx;
        const unsigned hx1 = (unsigned)(int)cx;
        const unsigned hy0 = (unsigned)(int)fy * P1;
        const unsigned hy1 = (unsigned)(int)cy * P1;
        const unsigned hz0 = (unsigned)(int)fz * P2;
        const unsigned hz1 = (unsigned)(int)cz * P2;

        float c0 = 0.0f, c1 = 0.0f;
        #pragma unroll
        for (int k = 0; k < 8; ++k) {                     // corner k = a*4+b*2+c, a->x, b->y, c->z
            const unsigned h = ((k & 4) ? hx1 : hx0)
                             ^ ((k & 2) ? hy1 : hy0)
                             ^ ((k & 1) ? hz1 : hz0);
            const float w = ((k & 4) ? dx : ex)
                          * ((k & 2) ? dy : ey)
                          * ((k & 1) ? dz : ez);
            const float2 f = tbl[h & (T_SIZE - 1)];       // ds_load_b64 gather from LDS
            c0 = fmaf(w, f.x, c0);
            c1 = fmaf(w, f.y, c1);
        }

        // out[n, level*2 : level*2+2] as one 8-byte store
        ((float2*)out)[(size_t)n * L_LEVELS + level] = make_float2(c0, c1);
    }
}

extern "C" void kernel_launch(void* const* d_in, const int* in_sizes, int n_in,
                              void* d_out, int out_size, void* d_ws, size_t ws_size,
                              hipStream_t stream) {
    const float* x   = (const float*)d_in[0];             // (262144, 3) f32
    const float* emb = (const float*)d_in[1];             // (16, 16384, 2) f32
    float* out = (float*)d_out;                           // (262144, 32) f32

    dim3 grid(L_LEVELS * CHUNKS);                         // 1024 blocks: (level, chunk)
    hash_encode_kernel<<<grid, THREADS, 0, stream>>>(x, emb, out);
}